// QuantumClassifier_amplitude_24756191494222
// MI455X (gfx1250) — compile-verified
//
#include <hip/hip_runtime.h>
#include <math.h>

typedef __attribute__((ext_vector_type(2))) float v2f;
typedef __attribute__((ext_vector_type(8))) float v8f;

#define MODES 10
#define PHOT 5
#define NSUB 32
#define K1 252
#define K2 2002
#define INPUT_DIM 784
#define FPI 3.14159265358979323846f

// ---------------------------------------------------------------------------
// Kernel 1: theta = (x @ W_down^T + b_down) / pi   via V_WMMA_F32_16X16X4_F32
// Grid: 16 blocks (16 rows each), Block: 32 threads (one wave = one 16x16 tile)
// W_down staged in LDS zero-padded to 16 rows -> unconditional ds_load_b64,
// no exec-mask predication in the K loop.
// ---------------------------------------------------------------------------
__global__ __launch_bounds__(32)
void down_proj_wmma(const float* __restrict__ x, const float* __restrict__ Wd,
                    const float* __restrict__ bd, float* __restrict__ theta) {
    __shared__ float Wlds[16 * INPUT_DIM];      // 50 KB, rows 10..15 zeroed
    const int lane = threadIdx.x;               // 0..31 (wave32)
    for (int i = lane; i < MODES * INPUT_DIM; i += 32) Wlds[i] = Wd[i];
    for (int i = lane; i < (16 - MODES) * INPUT_DIM; i += 32)
        Wlds[MODES * INPUT_DIM + i] = 0.0f;
    __syncthreads();

    const int row0 = blockIdx.x * 16;
    const int m  = lane & 15;   // M (for A) / N (for B,C) sub-index
    const int kh = lane >> 4;   // 0: K pair {0,1}; 1: K pair {2,3}

    v8f c = {};
    const float* xrow = x + (size_t)(row0 + m) * INPUT_DIM + 2 * kh;
    const float* wrow = Wlds + m * INPUT_DIM + 2 * kh;

    for (int k0 = 0; k0 < INPUT_DIM; k0 += 4) {
        // A (16x4 f32): lanes 0-15 -> M=lane, K={k0,k0+1}; lanes 16-31 -> K={k0+2,k0+3}
        const float2 av = *(const float2*)(xrow + k0);   // global_load_b64
        // B (4x16 f32): B[k][n] = W_down[n][k]; lanes hold N=lane&15, K pair by kh
        const float2 bv = *(const float2*)(wrow + k0);   // ds_load_b64
        v2f a; a.x = av.x; a.y = av.y;
        v2f b; b.x = bv.x; b.y = bv.y;
        // 8 args: (neg_a, A, neg_b, B, c_mod, C, reuse_a, reuse_b)
        c = __builtin_amdgcn_wmma_f32_16x16x4_f32(
                false, a, false, b, (short)0, c, false, false);
    }

    // C layout: lane<16 -> (M=j, N=lane); lane>=16 -> (M=j+8, N=lane-16)
    const int N = m;
    if (N < MODES) {
        const float bias = bd[N];
        #pragma unroll
        for (int j = 0; j < 8; ++j) {
            const int M = j + 8 * kh;
            theta[(size_t)(row0 + M) * MODES + N] = (c[j] + bias) * (1.0f / FPI);
        }
    }
}

// ---------------------------------------------------------------------------
// Shared circuit machinery: build R[s][m] = sum_{c in subset s} Ucols[m][c]
// where Ucols = UL @ diag(e^{i*theta}) @ UR[:, 0:10:2]   (10x5 complex)
// ---------------------------------------------------------------------------
__device__ __forceinline__ void build_R(const float* __restrict__ theta_n,
                                        const float* __restrict__ UL,
                                        const float* __restrict__ UR,
                                        float* __restrict__ Rre,   // LDS [32*10]
                                        float* __restrict__ Rim,   // LDS [32*10]
                                        float* __restrict__ sD,    // LDS [2*10]
                                        float* __restrict__ sUc) { // LDS [2*50]
    const int tid = threadIdx.x;
    if (tid < MODES) {
        const float th = theta_n[tid];
        sD[2 * tid + 0] = cosf(th);
        sD[2 * tid + 1] = sinf(th);
    }
    __syncthreads();
    if (tid < MODES * PHOT) {
        const int i  = tid / PHOT;
        const int cc = tid % PHOT;
        const int col = 2 * cc;                 // IN_MODES = 0,2,4,6,8
        float are = 0.f, aim = 0.f;
        #pragma unroll
        for (int t = 0; t < MODES; ++t) {
            const float ulr = UL[(i * MODES + t) * 2 + 0];
            const float uli = UL[(i * MODES + t) * 2 + 1];
            const float dre = sD[2 * t + 0];
            const float dim = sD[2 * t + 1];
            const float pre = ulr * dre - uli * dim;   // UL * D
            const float pim = ulr * dim + uli * dre;
            const float urr = UR[(t * MODES + col) * 2 + 0];
            const float uri = UR[(t * MODES + col) * 2 + 1];
            are += pre * urr - pim * uri;
            aim += pre * uri + pim * urr;
        }
        sUc[2 * tid + 0] = are;
        sUc[2 * tid + 1] = aim;
    }
    __syncthreads();
    for (int idx = tid; idx < NSUB * MODES; idx += blockDim.x) {
        const int s  = idx / MODES;
        const int mm = idx % MODES;
        float rre = 0.f, rim = 0.f;
        #pragma unroll
        for (int cc = 0; cc < PHOT; ++cc) {
            if ((s >> cc) & 1) {
                rre += sUc[(mm * PHOT + cc) * 2 + 0];
                rim += sUc[(mm * PHOT + cc) * 2 + 1];
            }
        }
        Rre[s * MODES + mm] = rre;
        Rim[s * MODES + mm] = rim;
    }
    __syncthreads();
}

// Ryser permanent |pm|^2 for a 5-row multiset, using precomputed R.
// SIGNS[s] = (-1)^(5 - popcount(s)) => (popcount odd) ? +1 : -1
__device__ __forceinline__ float perm_prob(const int* __restrict__ rowsk,
                                           const float* __restrict__ Rre,
                                           const float* __restrict__ Rim,
                                           int stagger) {
    const int m0 = rowsk[0], m1 = rowsk[1], m2 = rowsk[2], m3 = rowsk[3], m4 = rowsk[4];
    float pmr = 0.f, pmi = 0.f;
    #pragma unroll 4
    for (int ss = 0; ss < NSUB; ++ss) {
        const int s = (ss + stagger) & (NSUB - 1);   // stagger lanes -> spread LDS banks
        float ar = Rre[s * MODES + m0], ai = Rim[s * MODES + m0];
        float br = Rre[s * MODES + m1], bi = Rim[s * MODES + m1];
        float tr = ar * br - ai * bi, ti = ar * bi + ai * br;
        br = Rre[s * MODES + m2]; bi = Rim[s * MODES + m2];
        float ur = tr * br - ti * bi, ui = tr * bi + ti * br;
        br = Rre[s * MODES + m3]; bi = Rim[s * MODES + m3];
        tr = ur * br - ui * bi; ti = ur * bi + ui * br;
        br = Rre[s * MODES + m4]; bi = Rim[s * MODES + m4];
        ur = tr * br - ti * bi; ui = tr * bi + ti * br;
        const float sign = (__popc(s) & 1) ? 1.0f : -1.0f;
        pmr += sign * ur;
        pmi += sign * ui;
    }
    return pmr * pmr + pmi * pmi;
}

// ---------------------------------------------------------------------------
// Kernel 2: circuit 1 -> normalized p1 -> e1 (expected occupancies, 10 modes)
// One block (256 threads = 8 wave32s) per sample.
// ---------------------------------------------------------------------------
__global__ __launch_bounds__(256)
void circuit1_kernel(const float* __restrict__ theta, const float* __restrict__ U1L,
                     const float* __restrict__ U1R, const int* __restrict__ rows1,
                     float* __restrict__ e1out) {
    __shared__ float sD[2 * MODES];
    __shared__ float sUc[2 * MODES * PHOT];
    __shared__ float Rre[NSUB * MODES];
    __shared__ float Rim[NSUB * MODES];
    __shared__ float red[256];
    __shared__ float se1[MODES];

    const int n = blockIdx.x;
    const int tid = threadIdx.x;

    build_R(theta + (size_t)n * MODES, U1L, U1R, Rre, Rim, sD, sUc);

    float p = 0.f;
    if (tid < K1) p = perm_prob(rows1 + tid * PHOT, Rre, Rim, tid & 31);

    red[tid] = p;
    if (tid < MODES) se1[tid] = 0.f;
    __syncthreads();
    for (int off = 128; off > 0; off >>= 1) {
        if (tid < off) red[tid] += red[tid + off];
        __syncthreads();
    }
    const float inv = 1.0f / red[0];

    if (tid < K1) {
        const float pn = p * inv;
        #pragma unroll
        for (int r = 0; r < PHOT; ++r)
            atomicAdd(&se1[rows1[tid * PHOT + r]], pn);   // ds_add_f32
    }
    __syncthreads();
    if (tid < MODES) e1out[(size_t)n * MODES + tid] = se1[tid];
}

// ---------------------------------------------------------------------------
// Kernel 3: circuit 2 (theta = e1) -> pf -> fused out = pf @ W_out^T + b_out
// One block (256 threads) per sample; 2002 permanents in 8 strided passes.
// ---------------------------------------------------------------------------
__global__ __launch_bounds__(256)
void circuit2_kernel(const float* __restrict__ e1, const float* __restrict__ UfL,
                     const float* __restrict__ UfR, const int* __restrict__ rows2,
                     const float* __restrict__ norm2, const float* __restrict__ W_out,
                     const float* __restrict__ b_out, float* __restrict__ out) {
    __shared__ float sD[2 * MODES];
    __shared__ float sUc[2 * MODES * PHOT];
    __shared__ float Rre[NSUB * MODES];
    __shared__ float Rim[NSUB * MODES];
    __shared__ float red0[256];
    __shared__ float red1[256];

    const int n = blockIdx.x;
    const int tid = threadIdx.x;

    build_R(e1 + (size_t)n * MODES, UfL, UfR, Rre, Rim, sD, sUc);

    float a0 = 0.f, a1 = 0.f;
    for (int k = tid; k < K2; k += 256) {
        const float p = perm_prob(rows2 + k * PHOT, Rre, Rim, tid & 31) * norm2[k];
        a0 += p * W_out[k];        // W_out[0][k]
        a1 += p * W_out[K2 + k];   // W_out[1][k]
    }
    red0[tid] = a0;
    red1[tid] = a1;
    __syncthreads();
    for (int off = 128; off > 0; off >>= 1) {
        if (tid < off) { red0[tid] += red0[tid + off]; red1[tid] += red1[tid + off]; }
        __syncthreads();
    }
    if (tid == 0) {
        out[(size_t)n * 2 + 0] = red0[0] + b_out[0];
        out[(size_t)n * 2 + 1] = red1[0] + b_out[1];
    }
}

// ---------------------------------------------------------------------------
extern "C" void kernel_launch(void* const* d_in, const int* in_sizes, int n_in,
                              void* d_out, int out_size, void* d_ws, size_t ws_size,
                              hipStream_t stream) {
    (void)in_sizes; (void)n_in; (void)out_size; (void)ws_size;
    const float* x      = (const float*)d_in[0];   // (256, 784)
    const float* W_down = (const float*)d_in[1];   // (10, 784)
    const float* b_down = (const float*)d_in[2];   // (10,)
    const float* U1L    = (const float*)d_in[3];   // (10,10,2)
    const float* U1R    = (const float*)d_in[4];
    const float* UfL    = (const float*)d_in[5];
    const float* UfR    = (const float*)d_in[6];
    const float* W_out  = (const float*)d_in[7];   // (2, 2002)
    const float* b_out  = (const float*)d_in[8];   // (2,)
    // d_in[9] = mask1 (unused: reconstructed from rows1)
    const float* norm2  = (const float*)d_in[10];  // (2002,)
    const int*   rows1  = (const int*)d_in[11];    // (252, 5)
    const int*   rows2  = (const int*)d_in[12];    // (2002, 5)
    float* out = (float*)d_out;                    // (256, 2)

    float* theta = (float*)d_ws;                   // 256*10 f32
    float* e1    = theta + 256 * MODES;            // 256*10 f32

    down_proj_wmma<<<16, 32, 0, stream>>>(x, W_down, b_down, theta);
    circuit1_kernel<<<256, 256, 0, stream>>>(theta, U1L, U1R, rows1, e1);
    circuit2_kernel<<<256, 256, 0, stream>>>(e1, UfL, UfR, rows2, norm2,
                                             W_out, b_out, out);
}